// URNNCell_38500086841659
// MI455X (gfx1250) — compile-verified
//
#include <hip/hip_runtime.h>
#include <hip/hip_bf16.h>
#include <math.h>

// ---------------------------------------------------------------------------
// Problem constants (from reference): B=8192, N_IN=512, N=2048
// ---------------------------------------------------------------------------
#define BDIM   8192
#define KDIM   512        // N_IN
#define NFFT   2048       // N
#define NOUT   4096       // 2*N  (GEMM output columns)
#define NT     512        // threads per row-kernel block (16 waves)
#define LOG2N  11

typedef float v2f __attribute__((ext_vector_type(2)));
typedef float v8f __attribute__((ext_vector_type(8)));

// ---------------------------------------------------------------------------
// Kernel A: C[b, j] = sum_k inputs[b,k] * w_ih[j,k]   (inputs @ w_ih.T)
// fp32 WMMA 16x16x4. Block = 256 threads (8 waves), block tile 64(M) x 128(N),
// K staged through LDS in chunks of 32.
// ---------------------------------------------------------------------------
#define TK 32

__global__ __launch_bounds__(256)
void gemm_wmma_f32(const float* __restrict__ A,   // BDIM x KDIM
                   const float* __restrict__ W,   // NOUT x KDIM (w_ih, row-major)
                   float* __restrict__ C)         // BDIM x NOUT
{
    __shared__ float As[64][TK + 1];
    __shared__ float Bs[128][TK + 1];

    const int tid  = threadIdx.x;
    const int lane = tid & 31;
    const int wave = tid >> 5;          // 0..7
    const int wm   = wave & 3;          // 16-row strip within block tile
    const int wn   = wave >> 2;         // 64-col strip within block tile

    const int blockM = blockIdx.x * 64;
    const int blockN = blockIdx.y * 128;

    v8f acc[4] = {};                    // 16x64 accumulator strip per wave

    const int n16   = lane & 15;        // N (or M) index within fragment
    const int khalf = (lane >> 4) << 1; // 0 for lanes 0-15, 2 for lanes 16-31

    for (int kb = 0; kb < KDIM; kb += TK) {
        // Stage A tile: 64 x 32 floats (8 per thread), K-contiguous loads.
        #pragma unroll
        for (int i = tid; i < 64 * TK; i += 256) {
            int r = i / TK, c = i % TK;
            As[r][c] = A[(size_t)(blockM + r) * KDIM + kb + c];
        }
        // Stage B tile: 128 x 32 floats (16 per thread).
        #pragma unroll
        for (int i = tid; i < 128 * TK; i += 256) {
            int r = i / TK, c = i % TK;
            Bs[r][c] = W[(size_t)(blockN + r) * KDIM + kb + c];
        }
        // Hint next B tile toward the caches (emits global_prefetch_b8).
        if (kb + TK < KDIM)
            __builtin_prefetch(&W[(size_t)(blockN + (tid >> 1)) * KDIM + kb + TK], 0, 1);
        __syncthreads();

        const int aRow = wm * 16;
        const int bCol = wn * 64;
        #pragma unroll
        for (int kk = 0; kk < TK; kk += 4) {
            // A fragment 16x4 (ISA layout): lanes0-15 M=lane K={kk,kk+1},
            //                               lanes16-31 M=lane-16 K={kk+2,kk+3}
            v2f af;
            af.x = As[aRow + n16][kk + khalf + 0];
            af.y = As[aRow + n16][kk + khalf + 1];
            #pragma unroll
            for (int t = 0; t < 4; ++t) {
                v2f bf;
                bf.x = Bs[bCol + t * 16 + n16][kk + khalf + 0];
                bf.y = Bs[bCol + t * 16 + n16][kk + khalf + 1];
                acc[t] = __builtin_amdgcn_wmma_f32_16x16x4_f32(
                    /*neg_a=*/false, af, /*neg_b=*/false, bf,
                    /*c_mod=*/(short)0, acc[t],
                    /*reuse_a=*/false, /*reuse_b=*/false);
            }
        }
        __syncthreads();
    }

    // C fragment layout: VGPR r -> (M=r, N=lane) for lanes 0-15,
    //                              (M=8+r, N=lane-16) for lanes 16-31.
    const int mAdd    = (lane >> 4) * 8;
    const int rowBase = blockM + wm * 16;
    const int colBase = blockN + wn * 64;
    #pragma unroll
    for (int t = 0; t < 4; ++t) {
        #pragma unroll
        for (int r = 0; r < 8; ++r) {
            C[(size_t)(rowBase + mAdd + r) * NOUT + colBase + t * 16 + n16] = acc[t][r];
        }
    }
}

// ---------------------------------------------------------------------------
// Kernel B: fused per-row pipeline. One block (512 threads) per row.
// All intermediates live in LDS: 2 x 2048 complex ping-pong buffers (32 KB)
// + 6 KB reduction scratch.
// ---------------------------------------------------------------------------

__device__ __forceinline__ void block_reduce3(float a, float b, float c,
                                              float* redA, float* redB, float* redC,
                                              float out[3])
{
    const int tid = threadIdx.x;
    redA[tid] = a; redB[tid] = b; redC[tid] = c;
    __syncthreads();
    #pragma unroll
    for (int s = NT / 2; s > 0; s >>= 1) {
        if (tid < s) {
            redA[tid] += redA[tid + s];
            redB[tid] += redB[tid + s];
            redC[tid] += redC[tid + s];
        }
        __syncthreads();
    }
    out[0] = redA[0]; out[1] = redB[0]; out[2] = redC[0];
    __syncthreads();   // protect scratch before reuse
}

// Householder reflect: z -= (2/(sum|v|^2+eps)) * (z . conj(v)) * v
__device__ __forceinline__ void reflect(float2* cur,
                                        const float* __restrict__ re,
                                        const float* __restrict__ im,
                                        float* redA, float* redB, float* redC)
{
    const int tid = threadIdx.x;
    float sre = 0.f, sim = 0.f, svv = 0.f;
    for (int j = tid; j < NFFT; j += NT) {
        float2 z = cur[j];
        float vr = re[j], vi = im[j];
        sre += z.x * vr + z.y * vi;       // Re(z * conj(v))
        sim += z.y * vr - z.x * vi;       // Im(z * conj(v))
        svv += vr * vr + vi * vi;
    }
    float sums[3];
    block_reduce3(sre, sim, svv, redA, redB, redC, sums);
    const float dre = sums[0], dim_ = sums[1];
    const float factor = 2.0f / (sums[2] + 1e-6f);
    for (int j = tid; j < NFFT; j += NT) {
        float2 z = cur[j];
        float vr = re[j], vi = im[j];
        z.x -= factor * (dre * vr - dim_ * vi);
        z.y -= factor * (dre * vi + dim_ * vr);
        cur[j] = z;
    }
    __syncthreads();
}

// Stockham DIF radix-2 FFT, self-sorting. sign = -1 forward, +1 inverse
// (inverse is unscaled; caller applies 1/N). Returns buffer holding result.
__device__ __forceinline__ float2* fft_stockham(float2* x, float2* y, float sign)
{
    const int tid = threadIdx.x;
    int ncur = NFFT, s = 1, logs = 0;
    while (ncur > 1) {
        const int m = ncur >> 1;
        const float theta0 = sign * 6.2831853071795864769f / (float)ncur;
        for (int i = tid; i < NFFT / 2; i += NT) {
            const int p = i >> logs;
            const int q = i & (s - 1);
            float2 a = x[q + s * p];
            float2 b = x[q + s * (p + m)];
            float sn, cs;
            __sincosf(theta0 * (float)p, &sn, &cs);
            float2 apb = make_float2(a.x + b.x, a.y + b.y);
            float2 amb = make_float2(a.x - b.x, a.y - b.y);
            y[q + s * (2 * p)]     = apb;
            y[q + s * (2 * p + 1)] = make_float2(amb.x * cs - amb.y * sn,
                                                 amb.x * sn + amb.y * cs);
        }
        __syncthreads();
        ncur = m; s <<= 1; ++logs;
        float2* t = x; x = y; y = t;
    }
    return x;
}

__global__ __launch_bounds__(NT)
void urnn_row(const float* __restrict__ state,   // BDIM x 2N
              const float* __restrict__ mul,     // BDIM x 2N  (inputs @ w_ih.T)
              const float* __restrict__ b_h,     // N
              const float* __restrict__ w1,
              const float* __restrict__ re1, const float* __restrict__ im1,
              const float* __restrict__ w2,
              const float* __restrict__ re2, const float* __restrict__ im2,
              const float* __restrict__ w3,
              const int*   __restrict__ perm,    // N (int32)
              float* __restrict__ out)           // 2 copies of BDIM x 2N
{
    __shared__ float2 bufA[NFFT];
    __shared__ float2 bufB[NFFT];
    __shared__ float  redA[NT];
    __shared__ float  redB[NT];
    __shared__ float  redC[NT];

    const int tid = threadIdx.x;
    const int b   = blockIdx.x;
    const float* srow = state + (size_t)b * 2 * NFFT;

    // z = e^{i w1} * state_c
    for (int j = tid; j < NFFT; j += NT) {
        float sr = srow[j], si = srow[j + NFFT];
        float sn, cs;
        __sincosf(w1[j], &sn, &cs);
        bufA[j] = make_float2(cs * sr - sn * si, cs * si + sn * sr);
    }
    __syncthreads();

    // z = FFT(z)
    float2* cur = fft_stockham(bufA, bufB, -1.0f);

    // reflect 1
    reflect(cur, re1, im1, redA, redB, redC);

    // permutation gather + rotate by e^{i w2}
    float2* oth = (cur == bufA) ? bufB : bufA;
    for (int j = tid; j < NFFT; j += NT) {
        float2 z = cur[perm[j]];
        float sn, cs;
        __sincosf(w2[j], &sn, &cs);
        oth[j] = make_float2(cs * z.x - sn * z.y, cs * z.y + sn * z.x);
    }
    __syncthreads();

    // z = IFFT(z)  (unscaled; 1/N folded into final rotation — reflect is
    // linear & homogeneous so scaling commutes past it)
    cur = oth;
    oth = (cur == bufA) ? bufB : bufA;
    cur = fft_stockham(cur, oth, +1.0f);

    // reflect 2
    reflect(cur, re2, im2, redA, redB, redC);

    // final: z *= e^{i w3} / N ; preact = in_c + z ; modReLU ; write twice
    const float invN = 1.0f / (float)NFFT;
    const float* mrow  = mul + (size_t)b * 2 * NFFT;
    float* orow0 = out + (size_t)b * 2 * NFFT;
    float* orow1 = out + (size_t)BDIM * 2 * NFFT + (size_t)b * 2 * NFFT;
    for (int j = tid; j < NFFT; j += NT) {
        float2 z = cur[j];
        float sn, cs;
        __sincosf(w3[j], &sn, &cs);
        float zr = invN * (cs * z.x - sn * z.y);
        float zi = invN * (cs * z.y + sn * z.x);
        float pr = mrow[j] + zr;
        float pi = mrow[j + NFFT] + zi;
        float nrm = sqrtf(pr * pr + pi * pi);
        float rel = nrm + b_h[j];
        rel = rel > 0.0f ? rel : 0.0f;
        float scale = rel / (nrm + 1e-6f);
        float nr = pr * scale, ni = pi * scale;
        orow0[j] = nr;  orow0[j + NFFT] = ni;
        orow1[j] = nr;  orow1[j + NFFT] = ni;
    }
}

// ---------------------------------------------------------------------------
// Launch
// ---------------------------------------------------------------------------
extern "C" void kernel_launch(void* const* d_in, const int* in_sizes, int n_in,
                              void* d_out, int out_size, void* d_ws, size_t ws_size,
                              hipStream_t stream)
{
    const float* inputs = (const float*)d_in[0];
    const float* state  = (const float*)d_in[1];
    const float* w_ih   = (const float*)d_in[2];
    const float* b_h    = (const float*)d_in[3];
    const float* w1     = (const float*)d_in[4];
    const float* re1    = (const float*)d_in[5];
    const float* im1    = (const float*)d_in[6];
    const float* w2     = (const float*)d_in[7];
    const float* re2    = (const float*)d_in[8];
    const float* im2    = (const float*)d_in[9];
    const float* w3     = (const float*)d_in[10];
    const int*   perm   = (const int*)d_in[11];

    float* mul = (float*)d_ws;            // BDIM x NOUT fp32 intermediate (128 MB)
    float* outp = (float*)d_out;

    dim3 gGrid(BDIM / 64, NOUT / 128);    // 128 x 32 blocks
    gemm_wmma_f32<<<gGrid, 256, 0, stream>>>(inputs, w_ih, mul);

    urnn_row<<<BDIM, NT, 0, stream>>>(state, mul, b_h,
                                      w1, re1, im1, w2, re2, im2, w3,
                                      perm, outp);
}